// TELIF_9423158247572
// MI455X (gfx1250) — compile-verified
//
#include <hip/hip_runtime.h>
#include <hip/hip_bf16.h>

// -----------------------------------------------------------------------------
// TE-LIF spiking-neuron scan for MI455X (gfx1250).
//
// Roofline: ~258 MB HBM traffic (tx read + ty write + TE), ~0.23 GFLOP
//           => memory-bound, ~11 us floor at 23.3 TB/s. Design = streaming:
//   * 1 thread / neuron (b,n): tx/ty coalesced 128B-per-wave at each t,
//     2048 wave32's in flight for latency hiding.
//   * non-temporal loads/stores on the 256 MB tx/ty streams so the hot 2 MB
//     TE table stays resident in the 192 MB L2.
//   * TE is [N,T] (stride-T per neuron): pre-transpose to TEt[T,N] in d_ws.
//     The transpose is routed through the WMMA unit (A x Identity on
//     V_WMMA_F32_16X16X4_F32): D = A exactly, but the A-fragment
//     (row-per-lane) -> D-fragment (row-across-lanes) layout change performs
//     the shuffle in-register; stores follow the D layout => transposed.
// -----------------------------------------------------------------------------

typedef __attribute__((ext_vector_type(2))) float v2f;
typedef __attribute__((ext_vector_type(8))) float v8f;

#define T_STEPS 512
#define NN      1024          // neurons per batch row
#define BB      64            // batch
#define BN      (BB * NN)     // 65536 independent neurons

#define REST      0.0f
#define DECAY     0.2f
#define THRESH    0.3f
#define BETA      0.02f

// ---------------------------------------------------------------------------
// Kernel 1: TE[N,T] -> TEt[T,N] transpose through the WMMA unit.
// One wave (32 lanes) per 16x16 tile. A-fragment of V_WMMA_F32_16X16X4_F32:
//   lane L holds A[M = L%16, K = 2*(L/16) + r] in VGPR r (r = 0,1)  (ISA 7.12.2)
// B-fragment (4x16, rows-across-lanes, mirror of A):
//   lane L, VGPR r holds B[K = 2*(L/16) + r, Ncol = L%16]
// Accumulating A_chunk x I_chunk over 4 K-chunks yields D == 16x16 tile in
// C/D layout: lane L, VGPR r = D[M = r + 8*(L/16), Ncol = L%16].
// ---------------------------------------------------------------------------
__global__ __launch_bounds__(32) void te_transpose_wmma(
    const float* __restrict__ te,   // [NN, T_STEPS]
    float* __restrict__ tet)        // [T_STEPS, NN]
{
    const int tile   = blockIdx.x;            // (N/16)*(T/16) tiles
    const int tTiles = T_STEPS / 16;
    const int t0     = (tile % tTiles) * 16;
    const int n0     = (tile / tTiles) * 16;

    const int lane = threadIdx.x;             // 0..31, wave32, EXEC all ones
    const int row  = lane & 15;
    const int half = lane >> 4;

    v8f acc = {};
#pragma unroll
    for (int kc = 0; kc < 4; ++kc) {
        // A chunk: TE[n0+row, t0 + kc*4 + 2*half + {0,1}] -- aligned float2
        const float2 a2 = *reinterpret_cast<const float2*>(
            te + (size_t)(n0 + row) * T_STEPS + t0 + kc * 4 + 2 * half);
        v2f a; a[0] = a2.x; a[1] = a2.y;
        // Identity chunk: B[kk, j] = (j == kc*4 + kk)
        v2f b;
        b[0] = (row == kc * 4 + 2 * half + 0) ? 1.0f : 0.0f;
        b[1] = (row == kc * 4 + 2 * half + 1) ? 1.0f : 0.0f;
        acc = __builtin_amdgcn_wmma_f32_16x16x4_f32(
            /*neg_a=*/false, a, /*neg_b=*/false, b,
            /*c_mod=*/(short)0, acc, /*reuse_a=*/false, /*reuse_b=*/false);
    }

    // D[M, j] = TE[n0+M, t0+j]  ->  TEt[(t0+j)*NN + (n0+M)], j = row.
#pragma unroll
    for (int r = 0; r < 8; ++r) {
        const int m = r + 8 * half;
        tet[(size_t)(t0 + row) * NN + (n0 + m)] = acc[r];
    }
}

// ---------------------------------------------------------------------------
// Kernel 2: the LIF scan. One thread per neuron, sequential over t.
// tx/ty streamed non-temporally; TEt read through L2 (broadcast across b).
// ---------------------------------------------------------------------------
__global__ __launch_bounds__(256) void telif_scan(
    const float* __restrict__ tx,   // [T, B, N]
    const float* __restrict__ tet,  // [T, N] transposed TE
    float* __restrict__ out)        // [T, B, N]
{
    const int idx = blockIdx.x * blockDim.x + threadIdx.x;  // 0..BN-1
    const int n   = idx & (NN - 1);

    float v  = REST;
    float y  = 0.0f;
    float th = THRESH;

    const float* xp = tx + idx;
    const float* tp = tet + n;
    float*       op = out + idx;

#pragma unroll 4
    for (int t = 0; t < T_STEPS; ++t) {
        // global_prefetch_b8 a few steps ahead of the NT read stream
        if ((t & 3) == 0 && t + 16 < T_STEPS)
            __builtin_prefetch(xp + (size_t)(t + 16) * BN, 0, 1);

        const float x  = __builtin_nontemporal_load(xp + (size_t)t * BN);
        const float te = tp[(size_t)t * NN];

        th = (th + v * te) - (th - THRESH) * BETA;
        v  = v * (DECAY * (1.0f - y)) + x;    // (1-y) is exactly 0 or 1
        y  = (v > th) ? 1.0f : 0.0f;

        __builtin_nontemporal_store(y, op + (size_t)t * BN);
    }
}

// Fallback when d_ws can't hold the 2 MB transposed TE: strided TE reads
// (same-address across the 64 batch rows, served by L2).
__global__ __launch_bounds__(256) void telif_scan_direct(
    const float* __restrict__ tx,
    const float* __restrict__ te,   // [N, T]
    float* __restrict__ out)
{
    const int idx = blockIdx.x * blockDim.x + threadIdx.x;
    const int n   = idx & (NN - 1);

    float v = REST, y = 0.0f, th = THRESH;
    const float* xp = tx + idx;
    const float* tp = te + (size_t)n * T_STEPS;
    float*       op = out + idx;

#pragma unroll 4
    for (int t = 0; t < T_STEPS; ++t) {
        const float x  = __builtin_nontemporal_load(xp + (size_t)t * BN);
        const float tv = tp[t];
        th = (th + v * tv) - (th - THRESH) * BETA;
        v  = v * (DECAY * (1.0f - y)) + x;
        y  = (v > th) ? 1.0f : 0.0f;
        __builtin_nontemporal_store(y, op + (size_t)t * BN);
    }
}

extern "C" void kernel_launch(void* const* d_in, const int* in_sizes, int n_in,
                              void* d_out, int out_size, void* d_ws, size_t ws_size,
                              hipStream_t stream) {
    const float* tx = (const float*)d_in[0];   // [T, B, N] f32
    const float* TE = (const float*)d_in[1];   // [N, T]    f32
    float* out      = (float*)d_out;           // [T, B, N] f32 spikes
    (void)in_sizes; (void)n_in; (void)out_size;

    const size_t tet_bytes = (size_t)T_STEPS * NN * sizeof(float);
    if (ws_size >= tet_bytes) {
        float* tet = (float*)d_ws;
        // (N/16)*(T/16) = 2048 one-wave tiles
        te_transpose_wmma<<<(NN / 16) * (T_STEPS / 16), 32, 0, stream>>>(TE, tet);
        telif_scan<<<BN / 256, 256, 0, stream>>>(tx, tet, out);
    } else {
        telif_scan_direct<<<BN / 256, 256, 0, stream>>>(tx, TE, out);
    }
}